// IntMultiPrecConv2d_78323023609877
// MI455X (gfx1250) — compile-verified
//
#include <hip/hip_runtime.h>
#include <hip/hip_bf16.h>
#include <math.h>

#define BATCH 32
#define CIN   128
#define COUT  256
#define IMG_H 56
#define IMG_W 56

#define CK    8          // cin chunk staged per iteration
#define NCHUNK (CIN / CK)
#define WROW  72         // weight LDS row stride = CK*9 (TDM-native packing)
#define XSTR  67         // input LDS row stride (66 used, +1 pad)

typedef __attribute__((ext_vector_type(2))) float v2f;
typedef __attribute__((ext_vector_type(8))) float v8f;

#ifndef __has_builtin
#define __has_builtin(x) 0
#endif

#if __has_builtin(__builtin_amdgcn_tensor_load_to_lds) && \
    __has_builtin(__builtin_amdgcn_s_wait_tensorcnt)
#define HAVE_TDM 1
#else
#define HAVE_TDM 0
#endif

#if __has_include(<hip/amd_detail/amd_gfx1250_TDM.h>)
#define TDM_SIX_ARGS 1      // therock-10.0 headers -> 6-arg builtin
#else
#define TDM_SIX_ARGS 0      // ROCm 7.2 -> 5-arg builtin
#endif

#if HAVE_TDM
typedef unsigned int tdm_v4u __attribute__((ext_vector_type(4)));
typedef int          tdm_v4i __attribute__((ext_vector_type(4)));
typedef int          tdm_v8i __attribute__((ext_vector_type(8)));

// One 2-D TDM tile load: 64 rows (couts) x 72 floats (CK cin x 9 taps,
// contiguous in OIHW), global row stride CIN*9 = 1152 floats, packed rows
// in LDS at `lds_byte_off`. Uniform operands only; EXEC-independent.
static __device__ __forceinline__ void tdm_load_weights(const float* gsrc,
                                                        unsigned lds_byte_off) {
    unsigned long long ga = (unsigned long long)(uintptr_t)gsrc;
    tdm_v4u g0;
    g0[0] = 1u;                                            // count=1, user desc
    g0[1] = lds_byte_off;                                  // LDS dest (bytes)
    g0[2] = (unsigned)ga;                                  // global_addr[31:0]
    g0[3] = ((unsigned)(ga >> 32) & 0x01FFFFFFu)           // global_addr[56:32]
            | 0x80000000u;                                 // type=2 ("image")
    tdm_v8i g1;
    g1[0] = 0x00020000;            // data_size=2 (4 bytes); no pad/iter/multicast
    g1[1] = (int)(72u << 16);      // tensor_dim0[15:0]=72 (bits 79:48 low half)
    g1[2] = (int)(64u << 16);      // tensor_dim0 hi=0 | tensor_dim1[15:0]=64
    g1[3] = (int)(72u << 16);      // tensor_dim1 hi=0 | tile_dim0=72
    g1[4] = 64;                    // tile_dim1=64 | tile_dim2=0
    g1[5] = (int)(CIN * 9);        // tensor_dim0_stride[31:0] = 1152
    g1[6] = 0;                     // stride hi | tensor_dim1_stride lo (unused 2D)
    g1[7] = 0;
    tdm_v4i z4 = {};
#if TDM_SIX_ARGS
    tdm_v8i z8 = {};
    __builtin_amdgcn_tensor_load_to_lds(g0, g1, z4, z4, z8, 0);
#else
    __builtin_amdgcn_tensor_load_to_lds(g0, g1, z4, z4, 0);
#endif
}
#endif // HAVE_TDM

// ---------------------------------------------------------------------------
// Prologue: collapse the two quantization branches into per-channel scale/bias
// ---------------------------------------------------------------------------
__global__ void __launch_bounds__(256)
precompute_scale_bias(const float* __restrict__ aw,      // (2, COUT)
                      const float* __restrict__ alpha2,
                      const float* __restrict__ b8_2,
                      const float* __restrict__ nb_2,
                      const float* __restrict__ nsh_2,
                      const float* __restrict__ alpha8,
                      const float* __restrict__ b16_8,
                      const float* __restrict__ nsh_8,
                      float* __restrict__ scale,
                      float* __restrict__ bias) {
    int c = blockIdx.x * blockDim.x + threadIdx.x;
    if (c >= COUT) return;
    bool sel1 = aw[COUT + c] > aw[c];   // jnp.argmax: first max wins
    float s, bv;
    if (!sel1) {
        float inv = exp2f(-nsh_2[c]);
        s  = alpha2[c] * inv;
        bv = b8_2[c] * exp2f(nb_2[c]) * inv;
    } else {
        float inv = exp2f(-nsh_8[c]);
        s  = alpha8[c] * inv;
        bv = alpha8[c] * b16_8[c] * inv;
    }
    scale[c] = s;
    bias[c]  = bv;
}

// ---------------------------------------------------------------------------
// Implicit-GEMM 3x3 conv, fp32 via V_WMMA_F32_16X16X4_F32.
// Weights staged by the Tensor Data Mover (double-buffered, one in-flight
// prefetch); input slab staged per-lane (halo + zero pad not TDM-expressible).
// Block: 256 threads (8 waves). Tile: 64 cout x 64 spatial (one image row).
// ---------------------------------------------------------------------------
__global__ void __launch_bounds__(256)
intconv2d_wmma_f32(const float* __restrict__ x,       // (B, CIN, H, W)
                   const float* __restrict__ weight,  // (COUT, CIN, 3, 3)
                   const float* __restrict__ scale,   // (COUT)
                   const float* __restrict__ bias,    // (COUT)
                   float* __restrict__ out)           // (B, COUT, H, W)
{
    __shared__ float lds_w[2 * 64 * WROW];    // double-buffered TDM dest
    __shared__ float lds_x[CK * 3 * XSTR];    // [cin][row 0..2][w 0..65]

    const int cout_base = blockIdx.x * 64;
    const int h         = blockIdx.y;
    const int b         = blockIdx.z;

    const int lane = threadIdx.x & 31;
    const int wv   = threadIdx.x >> 5;        // 0..7
    const int cg   = wv >> 1;                 // cout subtile 0..3
    const int ng   = wv & 1;                  // n group 0..1
    const int l16  = lane & 15;
    const int kbase = (lane >> 4) << 1;       // lanes 0-15 -> K{0,1}; 16-31 -> K{2,3}

    const float* wblk = weight + (size_t)cout_base * CIN * 9;

#if HAVE_TDM
    if (wv == 0) {                            // one wave drives the TDM
        tdm_load_weights(wblk + 0 * CK * 9, (unsigned)(size_t)&lds_w[0]);
        if (NCHUNK > 1)
            tdm_load_weights(wblk + 1 * CK * 9,
                             (unsigned)(size_t)&lds_w[64 * WROW]);
    }
#endif

    v8f acc0 = {};
    v8f acc1 = {};

    for (int chunk = 0; chunk < NCHUNK; ++chunk) {
        const int cin0 = chunk * CK;
        float* wbuf = &lds_w[(chunk & 1) * 64 * WROW];

#if !HAVE_TDM
        // Fallback: cooperative weight staging in TDM-native layout.
        for (int p = threadIdx.x; p < 64 * CK * 9; p += 256) {
            const int co = p / (CK * 9);
            const int r  = p % (CK * 9);
            wbuf[co * WROW + r] = wblk[(size_t)co * CIN * 9 + cin0 * 9 + r];
        }
#endif
        // ---- stage input slab: CK cin x 3 rows x 66 w (zero-padded halo) ----
        for (int i = threadIdx.x; i < CK * 3 * 66; i += 256) {
            const int ci  = i / 198;
            const int rem = i % 198;
            const int r   = rem / 66;
            const int wl  = rem % 66;
            const int hg  = h + r - 1;
            const int wg_ = wl - 1;
            float v = 0.0f;
            if ((unsigned)hg < (unsigned)IMG_H && (unsigned)wg_ < (unsigned)IMG_W)
                v = x[(((size_t)b * CIN + cin0 + ci) * IMG_H + hg) * IMG_W + wg_];
            lds_x[(ci * 3 + r) * XSTR + wl] = v;
        }

#if HAVE_TDM
        if (wv == 0) {
            // chunk's weights must be resident; next chunk may stay in flight
            if (chunk + 1 < NCHUNK) __builtin_amdgcn_s_wait_tensorcnt(1);
            else                    __builtin_amdgcn_s_wait_tensorcnt(0);
        }
#endif
        __syncthreads();   // weights (TDM) + input slab visible to all waves

        // ---- 9 taps x (CK/4) k-steps x 2 n-subtiles of f32 WMMA ----
        #pragma unroll
        for (int tap = 0; tap < 9; ++tap) {
            const int kh = tap / 3;
            const int kw = tap % 3;
            #pragma unroll
            for (int ck = 0; ck < CK / 4; ++ck) {
                const int kk = ck * 4 + kbase;
                v2f a;   // A layout in LDS: [cout][cin*9 + tap], stride WROW
                a.x = wbuf[(cg * 16 + l16) * WROW + kk * 9 + tap];
                a.y = wbuf[(cg * 16 + l16) * WROW + (kk + 1) * 9 + tap];
                {
                    const int n = ng * 32 + l16;
                    v2f bf;
                    bf.x = lds_x[((kk    ) * 3 + kh) * XSTR + n + kw];
                    bf.y = lds_x[((kk + 1) * 3 + kh) * XSTR + n + kw];
                    acc0 = __builtin_amdgcn_wmma_f32_16x16x4_f32(
                        false, a, false, bf, (short)0, acc0, false, false);
                }
                {
                    const int n = ng * 32 + 16 + l16;
                    v2f bf;
                    bf.x = lds_x[((kk    ) * 3 + kh) * XSTR + n + kw];
                    bf.y = lds_x[((kk + 1) * 3 + kh) * XSTR + n + kw];
                    acc1 = __builtin_amdgcn_wmma_f32_16x16x4_f32(
                        false, a, false, bf, (short)0, acc1, false, false);
                }
            }
        }
        __syncthreads();   // all waves done with wbuf and lds_x

#if HAVE_TDM
        if (wv == 0 && chunk + 2 < NCHUNK)   // buffer (chunk&1) is now free
            tdm_load_weights(wblk + (chunk + 2) * CK * 9,
                             (unsigned)(size_t)&lds_w[(chunk & 1) * 64 * WROW]);
#endif
    }

    // ---- epilogue: out = scale[c]*acc + bias[c], masked to w < 56 ----
    const int mhalf = (lane >> 4) << 3;      // C/D: VGPR r -> M = r or r+8
    float sc[8], bs[8];
    #pragma unroll
    for (int r = 0; r < 8; ++r) {
        const int c = cout_base + cg * 16 + mhalf + r;
        sc[r] = scale[c];
        bs[r] = bias[c];
    }
    const int n0 = ng * 32 + l16;
    if (n0 < IMG_W) {
        #pragma unroll
        for (int r = 0; r < 8; ++r) {
            const int c = cout_base + cg * 16 + mhalf + r;
            out[(((size_t)b * COUT + c) * IMG_H + h) * IMG_W + n0] =
                sc[r] * acc0[r] + bs[r];
        }
    }
    const int n1 = n0 + 16;
    if (n1 < IMG_W) {
        #pragma unroll
        for (int r = 0; r < 8; ++r) {
            const int c = cout_base + cg * 16 + mhalf + r;
            out[(((size_t)b * COUT + c) * IMG_H + h) * IMG_W + n1] =
                sc[r] * acc1[r] + bs[r];
        }
    }
}

// ---------------------------------------------------------------------------
extern "C" void kernel_launch(void* const* d_in, const int* in_sizes, int n_in,
                              void* d_out, int out_size, void* d_ws, size_t ws_size,
                              hipStream_t stream) {
    const float* x      = (const float*)d_in[0];
    const float* weight = (const float*)d_in[1];
    const float* aw     = (const float*)d_in[2];
    const float* alpha2 = (const float*)d_in[3];
    const float* b8_2   = (const float*)d_in[4];
    const float* nb_2   = (const float*)d_in[5];
    const float* nsh_2  = (const float*)d_in[6];
    const float* alpha8 = (const float*)d_in[7];
    const float* b16_8  = (const float*)d_in[8];
    const float* nsh_8  = (const float*)d_in[9];
    float* out = (float*)d_out;

    float* scale = (float*)d_ws;          // COUT floats
    float* bias  = scale + COUT;          // COUT floats

    precompute_scale_bias<<<1, 256, 0, stream>>>(aw, alpha2, b8_2, nb_2, nsh_2,
                                                 alpha8, b16_8, nsh_8, scale, bias);

    dim3 grid(COUT / 64, IMG_H, BATCH);   // (4, 56, 32)
    intconv2d_wmma_f32<<<grid, 256, 0, stream>>>(x, weight, scale, bias, out);
}